// MyModel_60249801228295
// MI455X (gfx1250) — compile-verified
//
#include <hip/hip_runtime.h>
#include <math.h>

// Problem constants (match reference)
static constexpr int B = 256;
static constexpr int N = 4096;
static constexpr int E = 65536;
static constexpr int NC = 512;          // chunks for deterministic CSR build
static constexpr int CHUNK_E = E / NC;  // 128 edges per chunk

// Workspace byte offsets
static constexpr size_t WS_DEG_OUT = 0;                 // int[4096]
static constexpr size_t WS_DEG_IN  = 16384;             // int[4096]
static constexpr size_t WS_ROWPTR  = 32768;             // int[4097]
static constexpr size_t WS_CSRC    = 65536;             // float[4096]
static constexpr size_t WS_CDST    = 81920;             // float[4096]
static constexpr size_t WS_WT      = 98304;             // float[64*4]
static constexpr size_t WS_PART    = 131072;            // float[256*256]
static constexpr size_t WS_HIST    = 524288;            // int[4096*512]  (8 MB)
static constexpr size_t WS_CSR     = 9437184;           // int[65536]
static constexpr size_t WS_XT      = 10485760;          // float[4096*768] (12.6 MB)
static constexpr size_t WS_AGG4    = 25165824;          // float[256*4096*4] (16 MB)

typedef __attribute__((ext_vector_type(2))) float v2f;
typedef __attribute__((ext_vector_type(8))) float v8f;

// ---------------- K0: zero counters/hist ----------------
__global__ void k_init(int* __restrict__ degs, int* __restrict__ hist) {
    int i = blockIdx.x * blockDim.x + threadIdx.x;
    if (i < 2 * N) degs[i] = 0;
    int stride = gridDim.x * blockDim.x;
    for (int j = i; j < N * NC; j += stride) hist[j] = 0;
}

// ---------------- K0b: build Wt[64][4] = (W1^T | b1) ----------------
__global__ void k_wt(const float* __restrict__ W1, const float* __restrict__ b1,
                     float* __restrict__ Wt) {
    int k = threadIdx.x;
    if (k < 64) {
        Wt[k * 4 + 0] = W1[0 * 64 + k];
        Wt[k * 4 + 1] = W1[1 * 64 + k];
        Wt[k * 4 + 2] = W1[2 * 64 + k];
        Wt[k * 4 + 3] = b1[k];
    }
}

// ---------------- K1: degree counts + per-chunk dst histograms ----------------
__global__ void k_count(const int* __restrict__ src, const int* __restrict__ dst,
                        int* __restrict__ deg_out, int* __restrict__ deg_in,
                        int* __restrict__ hist) {
    int e = blockIdx.x * blockDim.x + threadIdx.x;
    if (e >= E) return;
    int s = src[e], d = dst[e];
    atomicAdd(&deg_out[s], 1);
    atomicAdd(&deg_in[d], 1);
    int chunk = e / CHUNK_E;
    atomicAdd(&hist[d * NC + chunk], 1);
}

// ---------------- K2a: exclusive scan of deg_in -> row_ptr (1 block, 1024 thr) ----
__global__ void k_scan(const int* __restrict__ deg_in, int* __restrict__ row_ptr) {
    __shared__ int lds[1024];
    int t = threadIdx.x;
    int vals[4];
    int tot = 0;
#pragma unroll
    for (int j = 0; j < 4; ++j) { vals[j] = deg_in[t * 4 + j]; tot += vals[j]; }
    lds[t] = tot;
    __syncthreads();
    for (int off = 1; off < 1024; off <<= 1) {
        int v = lds[t];
        int add = (t >= off) ? lds[t - off] : 0;
        __syncthreads();
        lds[t] = v + add;
        __syncthreads();
    }
    int run = lds[t] - tot;  // exclusive prefix for this thread's group
#pragma unroll
    for (int j = 0; j < 4; ++j) { row_ptr[t * 4 + j] = run; run += vals[j]; }
    if (t == 1023) row_ptr[N] = run;  // == E
}

// ---------------- K2b: norms + convert hist counts -> chunk offsets ------------
__global__ void k_offsets(const int* __restrict__ deg_out, const int* __restrict__ deg_in,
                          const int* __restrict__ row_ptr,
                          float* __restrict__ c_src, float* __restrict__ c_dst,
                          int* __restrict__ hist) {
    int n = blockIdx.x * blockDim.x + threadIdx.x;
    if (n >= N) return;
    float dout = (float)deg_out[n];
    float din  = (float)deg_in[n];
    c_src[n] = rsqrtf(dout > 1.0f ? dout : 1.0f);
    c_dst[n] = rsqrtf(din  > 1.0f ? din  : 1.0f);
    int base = row_ptr[n];
    for (int c = 0; c < NC; ++c) {
        int cnt = hist[n * NC + c];
        hist[n * NC + c] = base;
        base += cnt;
    }
}

// ---------------- K3: deterministic stable placement into CSR ----------------
__global__ void k_place(const int* __restrict__ src, const int* __restrict__ dst,
                        int* __restrict__ hist, int* __restrict__ csr_src) {
    int c = blockIdx.x * blockDim.x + threadIdx.x;
    if (c >= NC) return;
    int e0 = c * CHUNK_E;
    for (int i = 0; i < CHUNK_E; ++i) {
        int e = e0 + i;
        int d = dst[e];
        int pos = hist[d * NC + c];
        hist[d * NC + c] = pos + 1;  // slot (d,c) private to this thread
        csr_src[pos] = src[e];
    }
}

// ---------------- K4: transpose + source-scale: xt[n][b*3+j] ----------------
__global__ void k_xt(const float* __restrict__ x, const float* __restrict__ c_src,
                     float* __restrict__ xt) {
    size_t i = (size_t)blockIdx.x * blockDim.x + threadIdx.x;  // N*768 elements
    if (i >= (size_t)N * 768) return;
    int n = (int)(i / 768);
    int q = (int)(i % 768);
    int b = q / 3;
    int j = q - b * 3;
    xt[i] = x[((size_t)b * N + n) * 3 + j] * c_src[n];
}

// ---------------- K5: CSR SpMM aggregate -> agg4[b][n][4] (pad=1.0) ----------
__global__ void k_agg(const int* __restrict__ row_ptr, const int* __restrict__ csr_src,
                      const float* __restrict__ xt, const float* __restrict__ c_dst,
                      float* __restrict__ agg4) {
    int n = blockIdx.x;       // node
    int q = threadIdx.x;      // graph index b (256 threads, 3 floats each)
    int beg = row_ptr[n], end = row_ptr[n + 1];
    float a0 = 0.f, a1 = 0.f, a2 = 0.f;
    for (int idx = beg; idx < end; ++idx) {
        int s = csr_src[idx];
        const float* p = xt + (size_t)s * 768 + q * 3;
        a0 += p[0];
        a1 += p[1];
        a2 += p[2];
    }
    float cd = c_dst[n];
    float* o = agg4 + ((size_t)q * N + n) * 4;
    o[0] = a0 * cd;
    o[1] = a1 * cd;
    o[2] = a2 * cd;
    o[3] = 1.0f;  // K=4 pad row -> bias via Wt row 3
}

// ---------------- K6: WMMA [16x4]x[4x64] + fused ReLU + Wfc reduction --------
__global__ void __launch_bounds__(256)
k_wmma(const float* __restrict__ agg4, const float* __restrict__ Wt,
       const float* __restrict__ Wfc, float* __restrict__ partial) {
    int tid  = threadIdx.x;
    int lane = tid & 31;
    int wave = tid >> 5;
    int w    = blockIdx.x * 8 + wave;   // global wave id: 0..65535
    int b    = w >> 8;                  // graph
    int rt   = w & 255;                 // row tile
    int n0   = rt << 4;
    int half = lane >> 4;               // 0: K0/K1  1: K2/K3
    int lr   = lane & 15;

    // B tiles (4x16 each, 4 col tiles): VGPR0 = K(2*half), VGPR1 = K(2*half+1)
    v2f bt[4];
#pragma unroll
    for (int t = 0; t < 4; ++t) {
        const float* wp = Wt + ((t * 16 + lr) * 4 + half * 2);
        bt[t] = *(const v2f*)wp;
    }

    // A tile (16x4): row n0+lr, VGPR0 = K(2*half), VGPR1 = K(2*half+1)
    const float* ap = agg4 + (((size_t)b * N) + n0 + lr) * 4 + half * 2;
    v2f a = *(const v2f*)ap;

    float sum = 0.f;
#pragma unroll
    for (int t = 0; t < 4; ++t) {
        v8f c = {};
        c = __builtin_amdgcn_wmma_f32_16x16x4_f32(
                /*neg_a=*/false, a, /*neg_b=*/false, bt[t],
                /*c_mod=*/(short)0, c, /*reuse_a=*/false, /*reuse_b=*/false);
        // C layout: VGPR v, lanes0-15 -> M=v ; lanes16-31 -> M=v+8 ; N=lr
        int col = t * 16 + lr;
        const float* wf = Wfc + (size_t)(n0 + half * 8) * 64 + col;
#pragma unroll
        for (int v = 0; v < 8; ++v) {
            float hv = c[v] > 0.f ? c[v] : 0.f;      // ReLU
            sum += hv * wf[(size_t)v * 64];          // fused Wfc dot
        }
    }
    // wave32 reduction
#pragma unroll
    for (int off = 16; off > 0; off >>= 1) sum += __shfl_xor(sum, off, 32);
    if (lane == 0) partial[w] = sum;
}

// ---------------- K7: per-graph reduce + sigmoid ----------------
__global__ void k_final(const float* __restrict__ partial, const float* __restrict__ bfc,
                        float* __restrict__ out) {
    __shared__ float lds[256];
    int b = blockIdx.x;
    int t = threadIdx.x;
    lds[t] = partial[b * 256 + t];
    __syncthreads();
    for (int s = 128; s > 0; s >>= 1) {
        if (t < s) lds[t] += lds[t + s];
        __syncthreads();
    }
    if (t == 0) {
        float z = lds[0] + bfc[0];
        out[b] = 1.0f / (1.0f + expf(-z));
    }
}

extern "C" void kernel_launch(void* const* d_in, const int* in_sizes, int n_in,
                              void* d_out, int out_size, void* d_ws, size_t ws_size,
                              hipStream_t stream) {
    const float* x   = (const float*)d_in[0];
    const float* W1  = (const float*)d_in[1];
    const float* b1  = (const float*)d_in[2];
    const float* Wfc = (const float*)d_in[3];
    const float* bfc = (const float*)d_in[4];
    const int*   src = (const int*)d_in[5];
    const int*   dst = (const int*)d_in[6];
    float* out = (float*)d_out;

    char* ws = (char*)d_ws;
    int*   deg_out = (int*)(ws + WS_DEG_OUT);
    int*   deg_in  = (int*)(ws + WS_DEG_IN);
    int*   row_ptr = (int*)(ws + WS_ROWPTR);
    float* c_src   = (float*)(ws + WS_CSRC);
    float* c_dst   = (float*)(ws + WS_CDST);
    float* Wt      = (float*)(ws + WS_WT);
    float* part    = (float*)(ws + WS_PART);
    int*   hist    = (int*)(ws + WS_HIST);
    int*   csr     = (int*)(ws + WS_CSR);
    float* xt      = (float*)(ws + WS_XT);
    float* agg4    = (float*)(ws + WS_AGG4);

    k_init<<<2048, 256, 0, stream>>>(deg_out, hist);          // zeros both deg arrays (contiguous) + hist
    k_wt<<<1, 64, 0, stream>>>(W1, b1, Wt);
    k_count<<<E / 256, 256, 0, stream>>>(src, dst, deg_out, deg_in, hist);
    k_scan<<<1, 1024, 0, stream>>>(deg_in, row_ptr);
    k_offsets<<<N / 256, 256, 0, stream>>>(deg_out, deg_in, row_ptr, c_src, c_dst, hist);
    k_place<<<NC / 256, 256, 0, stream>>>(src, dst, hist, csr);
    k_xt<<<(N * 768 + 255) / 256, 256, 0, stream>>>(x, c_src, xt);
    k_agg<<<N, 256, 0, stream>>>(row_ptr, csr, xt, c_dst, agg4);
    k_wmma<<<(B * (N / 16)) / 8, 256, 0, stream>>>(agg4, Wt, Wfc, part);
    k_final<<<B, 256, 0, stream>>>(part, bfc, out);
}